// GinNet_64991445123423
// MI455X (gfx1250) — compile-verified
//
#include <hip/hip_runtime.h>
#include <hip/hip_bf16.h>

#define N_NODES 100000
#define N_EDGES 1200000
#define N_GRAPHS 256
#define HID 64
#define ROW_TILES (N_NODES / 16)   // 6250, exact

// LDS stride (in floats) between K-pairs of the swizzled W tile.
// 160 dwords => half-wave offset = 32 banks (mod 64): conflict-free b64 reads.
#define WSTRIDE 160

typedef float v2f __attribute__((ext_vector_type(2)));
typedef float v8f __attribute__((ext_vector_type(8)));

// ---------------------------------------------------------------- utilities
__global__ void zero_kernel(float* __restrict__ p, int n) {
    int i = blockIdx.x * 256 + threadIdx.x;
    if (i < n) p[i] = 0.0f;
}

// ------------------------------------------------------- edge aggregation
// layer 1: 3-wide features
__global__ void agg3_kernel(const float* __restrict__ x, const int* __restrict__ ei,
                            float* __restrict__ agg) {
    int e = blockIdx.x * 256 + threadIdx.x;
    if (e >= N_EDGES) return;
    int s = ei[e];
    int d = ei[N_EDGES + e];
    atomicAdd(&agg[d * 3 + 0], x[s * 3 + 0]);
    atomicAdd(&agg[d * 3 + 1], x[s * 3 + 1]);
    atomicAdd(&agg[d * 3 + 2], x[s * 3 + 2]);
}

// layers 2/3: 64-wide features; one thread handles a float4 chunk of one edge
__global__ void agg64_kernel(const float* __restrict__ h, const int* __restrict__ ei,
                             float* __restrict__ agg) {
    int i = blockIdx.x * 256 + threadIdx.x;      // edge*16 + chunk
    if (i >= N_EDGES * 16) return;
    int e = i >> 4;
    int c = (i & 15) << 2;
    int s = ei[e];
    int d = ei[N_EDGES + e];
    float4 v = *(const float4*)(h + s * 64 + c);
    float* dst = agg + d * 64 + c;
    atomicAdd(dst + 0, v.x);
    atomicAdd(dst + 1, v.y);
    atomicAdd(dst + 2, v.z);
    atomicAdd(dst + 3, v.w);
}

// --------------------------------------------- layer-1 first linear (3->64)
__global__ void lin3_kernel(const float* __restrict__ x, const float* __restrict__ agg3,
                            const float* __restrict__ epsp,
                            const float* __restrict__ W, const float* __restrict__ b,
                            float* __restrict__ out) {
    int i = blockIdx.x * 256 + threadIdx.x;      // node*64 + f
    if (i >= N_NODES * 64) return;
    int n = i >> 6, f = i & 63;
    float s = 1.0f + epsp[0];
    float t0 = s * x[n * 3 + 0] + agg3[n * 3 + 0];
    float t1 = s * x[n * 3 + 1] + agg3[n * 3 + 1];
    float t2 = s * x[n * 3 + 2] + agg3[n * 3 + 2];
    float v = t0 * W[0 * 64 + f] + t1 * W[1 * 64 + f] + t2 * W[2 * 64 + f] + b[f];
    out[i] = v > 0.0f ? v : 0.0f;
}

// ------------------------------------------------------- WMMA GEMM (Nx64 @ 64x64)
// One wave32 computes a 16x64 output tile: 4 col-tiles, K=64 in 16 steps of
// v_wmma_f32_16x16x4_f32.  W is staged in LDS pre-swizzled into B-fragment
// order (K-pairs interleaved per column) so every fragment is a single
// aligned ds_load_b64.  All 16 A fragments are preloaded (GIN combine fused
// when FUSE).  Epilogue: + bias, ReLU.
template <bool FUSE>
__global__ __launch_bounds__(256) void gemm64_kernel(
    const float* __restrict__ in, const float* __restrict__ agg,
    const float* __restrict__ epsp,
    const float* __restrict__ W, const float* __restrict__ bias,
    float* __restrict__ out, int nTiles) {
    __shared__ float Ws[32 * WSTRIDE];   // 32 K-pairs x (64 cols x 2) padded
    __shared__ float bs[64];
    int tid = threadIdx.x;
    for (int i = tid; i < 64 * 64; i += 256) {
        int k = i >> 6, col = i & 63;
        Ws[(k >> 1) * WSTRIDE + col * 2 + (k & 1)] = W[i];
    }
    if (tid < 64) bs[tid] = bias[tid];
    __syncthreads();

    float s = 1.0f;
    if constexpr (FUSE) s = 1.0f + epsp[0];

    int wave = tid >> 5;
    int lane = tid & 31;
    int tile = blockIdx.x * 8 + wave;
    if (tile >= nTiles) return;           // wave-uniform: EXEC stays all-1s

    int row0 = tile * 16;
    int half = lane >> 4;                 // which half-wave
    int l16  = lane & 15;

    // ---- preload all A fragments for this wave's 16 rows ----
    // A layout (16x4 f32): lanes 0-15 hold K={4k,4k+1}, lanes 16-31 K={4k+2,4k+3}
    const float* aPtr = in + (row0 + l16) * 64 + half * 2;
    v2f afrag[16];
    if constexpr (FUSE) {
        const float* gPtr = agg + (row0 + l16) * 64 + half * 2;
#pragma unroll
        for (int ks = 0; ks < 16; ++ks) {
            afrag[ks].x = s * aPtr[ks * 4]     + gPtr[ks * 4];
            afrag[ks].y = s * aPtr[ks * 4 + 1] + gPtr[ks * 4 + 1];
        }
    } else {
#pragma unroll
        for (int ks = 0; ks < 16; ++ks) {
            afrag[ks] = *(const v2f*)(aPtr + ks * 4);
        }
    }

    v8f acc0 = {}, acc1 = {}, acc2 = {}, acc3 = {};
#pragma unroll
    for (int ks = 0; ks < 16; ++ks) {
        // B fragment: lanes 0-15 hold K rows {4ks,4ks+1}, lanes 16-31 {4ks+2,4ks+3}
        const float* bbase = &Ws[(ks * 2 + half) * WSTRIDE + l16 * 2];
        v2f b0 = *(const v2f*)(bbase +  0);
        v2f b1 = *(const v2f*)(bbase + 32);
        v2f b2 = *(const v2f*)(bbase + 64);
        v2f b3 = *(const v2f*)(bbase + 96);
        acc0 = __builtin_amdgcn_wmma_f32_16x16x4_f32(false, afrag[ks], false, b0, (short)0, acc0, false, false);
        acc1 = __builtin_amdgcn_wmma_f32_16x16x4_f32(false, afrag[ks], false, b1, (short)0, acc1, false, false);
        acc2 = __builtin_amdgcn_wmma_f32_16x16x4_f32(false, afrag[ks], false, b2, (short)0, acc2, false, false);
        acc3 = __builtin_amdgcn_wmma_f32_16x16x4_f32(false, afrag[ks], false, b3, (short)0, acc3, false, false);
    }
    // C/D layout: VGPR j holds row M = j + 8*half, col = lane&15 (per 16-col tile)
#pragma unroll
    for (int j = 0; j < 8; ++j) {
        int r = row0 + j + 8 * half;
        float* op = out + r * 64 + l16;
        float v0 = acc0[j] + bs[l16 +  0]; op[0]  = v0 > 0.0f ? v0 : 0.0f;
        float v1 = acc1[j] + bs[l16 + 16]; op[16] = v1 > 0.0f ? v1 : 0.0f;
        float v2 = acc2[j] + bs[l16 + 32]; op[32] = v2 > 0.0f ? v2 : 0.0f;
        float v3 = acc3[j] + bs[l16 + 48]; op[48] = v3 > 0.0f ? v3 : 0.0f;
    }
}

// ------------------------------------------------------------- BatchNorm
// stride-256 walk => each thread stays on feature f = tid&63; local partials,
// two global atomics per thread.
__global__ void bn_stats_kernel(const float* __restrict__ h, float* __restrict__ stats, int n) {
    int tid = threadIdx.x;
    float lsum = 0.0f, lsq = 0.0f;
    for (int i = blockIdx.x * 256 + tid; i < n; i += gridDim.x * 256) {
        float v = h[i];
        lsum += v;
        lsq += v * v;
    }
    int f = tid & 63;
    atomicAdd(&stats[f], lsum);
    atomicAdd(&stats[64 + f], lsq);
}

__global__ void bn_apply_kernel(float* __restrict__ h, const float* __restrict__ stats,
                                const float* __restrict__ gm, const float* __restrict__ bt,
                                int n, float invN) {
    int i = blockIdx.x * 256 + threadIdx.x;
    if (i >= n) return;
    int f = i & 63;
    float m = stats[f] * invN;
    float var = stats[64 + f] * invN - m * m;
    float r = __frsqrt_rn(var + 1e-5f);
    h[i] = (h[i] - m) * r * gm[f] + bt[f];
}

// ------------------------------------------------------------- pooling
__global__ void pool_kernel(const float* __restrict__ h, const int* __restrict__ batch,
                            float* __restrict__ psum) {
    int i = blockIdx.x * 256 + threadIdx.x;
    if (i >= N_NODES * 64) return;
    int n = i >> 6, f = i & 63;
    atomicAdd(&psum[batch[n] * 64 + f], h[i]);
}

__global__ void cnt_kernel(const int* __restrict__ batch, float* __restrict__ pcnt) {
    int n = blockIdx.x * 256 + threadIdx.x;
    if (n >= N_NODES) return;
    atomicAdd(&pcnt[batch[n]], 1.0f);
}

// ---------------------------------------------- head: BN + FC1/ELU + FC2 + logsoftmax
__global__ __launch_bounds__(256) void head_kernel(
    const float* __restrict__ psum, const float* __restrict__ pcnt,
    const float* __restrict__ g3, const float* __restrict__ be3,
    const float* __restrict__ fc1w, const float* __restrict__ fc1b,
    const float* __restrict__ fc2w, const float* __restrict__ fc2b,
    float* __restrict__ out) {
    __shared__ float sm[64], sv[64];
    int tid = threadIdx.x;
    if (tid < 64) {
        float s = 0.0f, q = 0.0f;
        for (int g = 0; g < N_GRAPHS; ++g) {
            float c = pcnt[g]; c = c > 1.0f ? c : 1.0f;
            float p = psum[g * 64 + tid] / c;
            s += p; q += p * p;
        }
        float m = s * (1.0f / N_GRAPHS);
        float var = q * (1.0f / N_GRAPHS) - m * m;
        sm[tid] = m;
        sv[tid] = __frsqrt_rn(var + 1e-5f);
    }
    __syncthreads();
    int g = tid;  // one graph per thread
    float c = pcnt[g]; c = c > 1.0f ? c : 1.0f;
    float z[32];
#pragma unroll
    for (int j = 0; j < 32; ++j) z[j] = fc1b[j];
    for (int f = 0; f < 64; ++f) {
        float p = psum[g * 64 + f] / c;
        p = (p - sm[f]) * sv[f] * g3[f] + be3[f];
#pragma unroll
        for (int j = 0; j < 32; ++j) z[j] += p * fc1w[f * 32 + j];
    }
#pragma unroll
    for (int j = 0; j < 32; ++j) {
        float v = z[j];
        z[j] = v > 0.0f ? v : (__expf(v) - 1.0f);   // ELU
    }
    float o[10];
#pragma unroll
    for (int k = 0; k < 10; ++k) o[k] = fc2b[k];
    for (int j = 0; j < 32; ++j)
#pragma unroll
        for (int k = 0; k < 10; ++k) o[k] += z[j] * fc2w[j * 10 + k];
    float mx = o[0];
#pragma unroll
    for (int k = 1; k < 10; ++k) mx = o[k] > mx ? o[k] : mx;
    float se = 0.0f;
#pragma unroll
    for (int k = 0; k < 10; ++k) se += __expf(o[k] - mx);
    float lse = mx + __logf(se);
#pragma unroll
    for (int k = 0; k < 10; ++k) out[g * 10 + k] = o[k] - lse;
}

// ---------------------------------------------------------------- launch
extern "C" void kernel_launch(void* const* d_in, const int* in_sizes, int n_in,
                              void* d_out, int out_size, void* d_ws, size_t ws_size,
                              hipStream_t stream) {
    (void)in_sizes; (void)n_in; (void)out_size; (void)ws_size;
    const float* x     = (const float*)d_in[0];
    const int*   ei    = (const int*)d_in[1];
    const int*   batch = (const int*)d_in[2];
    const float* w1a = (const float*)d_in[3];  const float* b1a = (const float*)d_in[4];
    const float* w1b = (const float*)d_in[5];  const float* b1b = (const float*)d_in[6];
    const float* eps1= (const float*)d_in[7];
    const float* g1  = (const float*)d_in[8];  const float* be1 = (const float*)d_in[9];
    const float* w2a = (const float*)d_in[10]; const float* b2a = (const float*)d_in[11];
    const float* w2b = (const float*)d_in[12]; const float* b2b = (const float*)d_in[13];
    const float* eps2= (const float*)d_in[14];
    const float* g2  = (const float*)d_in[15]; const float* be2 = (const float*)d_in[16];
    const float* w3a = (const float*)d_in[17]; const float* b3a = (const float*)d_in[18];
    const float* w3b = (const float*)d_in[19]; const float* b3b = (const float*)d_in[20];
    const float* eps3= (const float*)d_in[21];
    const float* g3  = (const float*)d_in[22]; const float* be3 = (const float*)d_in[23];
    const float* fc1w= (const float*)d_in[24]; const float* fc1b= (const float*)d_in[25];
    const float* fc2w= (const float*)d_in[26]; const float* fc2b= (const float*)d_in[27];

    float* ws    = (float*)d_ws;
    float* h     = ws;                     // N_NODES*64
    float* u     = h + N_NODES * 64;       // N_NODES*64
    float* agg   = u + N_NODES * 64;       // N_NODES*64 (layer1 uses first 3*N)
    float* stats = agg + N_NODES * 64;     // 128
    float* psum  = stats + 128;            // 256*64
    float* pcnt  = psum + N_GRAPHS * 64;   // 256
    float* out   = (float*)d_out;

    const int TPB = 256;
    const int NF = N_NODES * 64;
    auto cdiv = [](int a, int b) { return (a + b - 1) / b; };
    const int gemmBlocks = cdiv(ROW_TILES, 8);

    // ---- layer 1 ----
    zero_kernel<<<cdiv(N_NODES * 3, TPB), TPB, 0, stream>>>(agg, N_NODES * 3);
    agg3_kernel<<<cdiv(N_EDGES, TPB), TPB, 0, stream>>>(x, ei, agg);
    lin3_kernel<<<cdiv(NF, TPB), TPB, 0, stream>>>(x, agg, eps1, w1a, b1a, u);
    gemm64_kernel<false><<<gemmBlocks, TPB, 0, stream>>>(u, nullptr, nullptr, w1b, b1b, h, ROW_TILES);
    zero_kernel<<<1, TPB, 0, stream>>>(stats, 128);
    bn_stats_kernel<<<2048, TPB, 0, stream>>>(h, stats, NF);
    bn_apply_kernel<<<cdiv(NF, TPB), TPB, 0, stream>>>(h, stats, g1, be1, NF, 1.0f / N_NODES);

    // ---- layer 2 ----
    zero_kernel<<<cdiv(NF, TPB), TPB, 0, stream>>>(agg, NF);
    agg64_kernel<<<cdiv(N_EDGES * 16, TPB), TPB, 0, stream>>>(h, ei, agg);
    gemm64_kernel<true><<<gemmBlocks, TPB, 0, stream>>>(h, agg, eps2, w2a, b2a, u, ROW_TILES);
    gemm64_kernel<false><<<gemmBlocks, TPB, 0, stream>>>(u, nullptr, nullptr, w2b, b2b, h, ROW_TILES);
    zero_kernel<<<1, TPB, 0, stream>>>(stats, 128);
    bn_stats_kernel<<<2048, TPB, 0, stream>>>(h, stats, NF);
    bn_apply_kernel<<<cdiv(NF, TPB), TPB, 0, stream>>>(h, stats, g2, be2, NF, 1.0f / N_NODES);

    // ---- layer 3 ----
    zero_kernel<<<cdiv(NF, TPB), TPB, 0, stream>>>(agg, NF);
    agg64_kernel<<<cdiv(N_EDGES * 16, TPB), TPB, 0, stream>>>(h, ei, agg);
    gemm64_kernel<true><<<gemmBlocks, TPB, 0, stream>>>(h, agg, eps3, w3a, b3a, u, ROW_TILES);
    gemm64_kernel<false><<<gemmBlocks, TPB, 0, stream>>>(u, nullptr, nullptr, w3b, b3b, h, ROW_TILES);

    // ---- pool + head ----
    zero_kernel<<<cdiv(N_GRAPHS * 64 + N_GRAPHS, TPB), TPB, 0, stream>>>(psum, N_GRAPHS * 64 + N_GRAPHS);
    pool_kernel<<<cdiv(NF, TPB), TPB, 0, stream>>>(h, batch, psum);
    cnt_kernel<<<cdiv(N_NODES, TPB), TPB, 0, stream>>>(batch, pcnt);
    head_kernel<<<1, TPB, 0, stream>>>(psum, pcnt, g3, be3, fc1w, fc1b, fc2w, fc2b, out);
}